// Encoder_16595753632230
// MI455X (gfx1250) — compile-verified
//
#include <hip/hip_runtime.h>

// Problem constants (fixed by the reference)
#define BB   2
#define TT   15
#define NN   2048
#define TO   8      // len(anchor_t) = arange(1,16,2)
#define MM   64     // N / SPATIAL_STRIDE
#define KK   32
#define DIMD 512
#define RAD2 0.49f  // RADIUS^2

typedef float v2f __attribute__((ext_vector_type(2)));
typedef float v8f __attribute__((ext_vector_type(8)));

// ---------------------------------------------------------------------------
// Kernel 1: farthest point sampling, one wave32 per (b, to) problem.
// Points staged in LDS as float4; per-lane mind[] kept in registers;
// argmax reduced with barrier-free shfl_xor butterflies (min-index tiebreak,
// matching jnp.argmax "first max" semantics).
// ---------------------------------------------------------------------------
__global__ __launch_bounds__(32) void fps_kernel(const float* __restrict__ input,
                                                 float4* __restrict__ anchors) {
  const int bt   = blockIdx.x;       // b*TO + to
  const int b    = bt / TO;
  const int to   = bt % TO;
  const int lane = threadIdx.x;      // wave32
  const int tsrc = 2 * to;           // anchor_t = 2*to+1 in padded time -> orig 2*to
  const float* fr = input + ((size_t)(b * TT + tsrc) * NN) * 3;

  __shared__ float4 pts[NN];         // 32 KB
  for (int i = lane; i < NN; i += 32) {
    pts[i] = make_float4(fr[3 * i], fr[3 * i + 1], fr[3 * i + 2], 0.f);
  }
  __syncthreads();

  float mind[NN / 32];
#pragma unroll
  for (int j = 0; j < NN / 32; ++j) mind[j] = 1e10f;

  float4* aout = anchors + (size_t)bt * MM;
  if (lane == 0) aout[0] = pts[0];

  int last = 0;
  for (int s = 1; s < MM; ++s) {
    const float4 lp = pts[last];     // uniform LDS broadcast
    float best = -1.0f;
    int   bi   = 0x7fffffff;
#pragma unroll 8
    for (int j = 0; j < NN / 32; ++j) {
      const int i = j * 32 + lane;
      float4 p = pts[i];
      float dx = p.x - lp.x, dy = p.y - lp.y, dz = p.z - lp.z;
      float d  = dx * dx + dy * dy + dz * dz;
      float mv = fminf(mind[j], d);
      mind[j] = mv;
      if (mv > best) { best = mv; bi = i; }   // strict > keeps earliest i per lane
    }
#pragma unroll
    for (int off = 16; off >= 1; off >>= 1) {
      float ob = __shfl_xor(best, off, 32);
      int   oi = __shfl_xor(bi, off, 32);
      if (ob > best || (ob == best && oi < bi)) { best = ob; bi = oi; }
    }
    last = bi;                        // uniform across lanes now
    if (lane == 0) aout[s] = pts[last];
  }
}

// ---------------------------------------------------------------------------
// Kernel 2: one block (4 waves) per anchor. Waves 0..2 do ball query + gather
// for dt = -1,0,+1 (wave32 ballot stream-compaction, matching the reference's
// "first K ascending indices" rule + first-hit padding). Then all 4 waves run
// the D=512 projection as V_WMMA_F32_16X16X4_F32 (K-dim is exactly 4), with a
// second chained WMMA folding the feat*W_f rank-1 term, max-pool over K and
// dt in the C-register layout, and the pos/ReLU epilogue.
// ---------------------------------------------------------------------------
__global__ __launch_bounds__(128) void encode_kernel(
    const float* __restrict__ input, const float* __restrict__ Wd,
    const float* __restrict__ Wf, const float* __restrict__ Wpos,
    const float* __restrict__ bpos, const float4* __restrict__ anchors,
    float* __restrict__ out) {
  const int aid  = blockIdx.x;             // ((b*TO)+to)*MM + m  == output row
  const int bt   = aid / MM;
  const int b    = bt / TO;
  const int to   = bt % TO;
  const int tid  = threadIdx.x;
  const int w    = tid >> 5;
  const int lane = tid & 31;

  __shared__ float As[3][KK][4];           // [dt][point][dx,dy,dz,dt]
  __shared__ float Gf[3][KK];              // gathered feat (= z)
  __shared__ int   lst[3][KK];

  const float4 anc = anchors[aid];

  if (w < 3) {
    const int dt = w - 1;
    int tp = 2 * to + 1 + dt;              // padded time 0..16
    int ts = tp - 1;
    if (ts < 0) ts = 0;
    if (ts > TT - 1) ts = TT - 1;
    const float* fr = input + ((size_t)(b * TT + ts) * NN) * 3;

    int cnt = 0;                           // uniform across the wave
    for (int base = 0; base < NN && cnt < KK; base += 32) {
      const int i = base + lane;
      float px = fr[3 * i], py = fr[3 * i + 1], pz = fr[3 * i + 2];
      float dx = px - anc.x, dy = py - anc.y, dz = pz - anc.z;
      float d2 = dx * dx + dy * dy + dz * dz;
      const bool pred = d2 < RAD2;
      const unsigned mask = (unsigned)__ballot((int)pred);  // wave32: low 32 bits
      const unsigned lt = (1u << lane) - 1u;
      const int pos = cnt + __popc(mask & lt);
      if (pred && pos < KK) lst[w][pos] = i;
      cnt += __popc(mask);
    }
    if (cnt > KK) cnt = KK;
    const int i0 = (cnt > 0) ? lst[w][0] : 0;   // reference: first==N -> 0
    if (lane < KK) {
      const int idx = (lane < cnt) ? lst[w][lane] : i0;
      float px = fr[3 * idx], py = fr[3 * idx + 1], pz = fr[3 * idx + 2];
      As[w][lane][0] = px - anc.x;
      As[w][lane][1] = py - anc.y;
      As[w][lane][2] = pz - anc.z;
      As[w][lane][3] = (float)dt;
      Gf[w][lane]    = pz;                 // feat channel is z
    }
  }
  __syncthreads();

  const int half = lane >> 4;              // 0: K=0,1  1: K=2,3 (A/B frag halves)
  const int l16  = lane & 15;

  // A fragments: lane<16 holds (M=l16, K=0/1), lane>=16 holds (M=l16, K=2/3)
  v2f afrag[3][2];
  v2f a2frag[3][2];
#pragma unroll
  for (int d3 = 0; d3 < 3; ++d3) {
#pragma unroll
    for (int r = 0; r < 2; ++r) {
      const int row = r * 16 + l16;
      v2f a; a.x = As[d3][row][2 * half]; a.y = As[d3][row][2 * half + 1];
      afrag[d3][r] = a;
      v2f a2; a2.x = (half == 0) ? Gf[d3][row] : 0.f; a2.y = 0.f;
      a2frag[d3][r] = a2;                  // rank-1 feat term: A2=[gf,0,0,0]
    }
  }

  const float tval = (float)(to + 1);
  float* orow = out + (size_t)aid * DIMD;

  // Each wave owns 8 column tiles of 16 dims -> 128 dims; 4 waves cover 512.
  for (int cl = 0; cl < 8; ++cl) {
    const int c  = w * 8 + cl;
    const int d0 = c * 16 + l16;
    // B = W_d^T (4 x 16): lane<16 holds (K=0/1, N=l16), lane>=16 (K=2/3, N=l16)
    v2f bfrag;  bfrag.x  = Wd[d0 * 4 + 2 * half]; bfrag.y = Wd[d0 * 4 + 2 * half + 1];
    v2f b2frag; b2frag.x = (half == 0) ? Wf[d0] : 0.f; b2frag.y = 0.f;

    float pooled = -3.402823466e38f;
#pragma unroll
    for (int d3 = 0; d3 < 3; ++d3) {
#pragma unroll
      for (int r = 0; r < 2; ++r) {
        v8f cacc = {};
        cacc = __builtin_amdgcn_wmma_f32_16x16x4_f32(
            false, afrag[d3][r], false, bfrag, (short)0, cacc, false, false);
        cacc = __builtin_amdgcn_wmma_f32_16x16x4_f32(
            false, a2frag[d3][r], false, b2frag, (short)0, cacc, false, false);
        // max over the 16 rows of this tile: 8 in-lane + fold lane halves
        float mx = cacc[0];
#pragma unroll
        for (int v = 1; v < 8; ++v) mx = fmaxf(mx, cacc[v]);
        mx = fmaxf(mx, __shfl_xor(mx, 16, 32));
        pooled = fmaxf(pooled, mx);
      }
    }
    // pos head + ReLU
    float pos = anc.x * Wpos[d0 * 4 + 0] + anc.y * Wpos[d0 * 4 + 1] +
                anc.z * Wpos[d0 * 4 + 2] + tval * Wpos[d0 * 4 + 3] + bpos[d0];
    float vout = pos + pooled;
    if (vout < 0.f) vout = 0.f;
    if (half == 0) orow[d0] = vout;
  }
}

extern "C" void kernel_launch(void* const* d_in, const int* in_sizes, int n_in,
                              void* d_out, int out_size, void* d_ws, size_t ws_size,
                              hipStream_t stream) {
  (void)in_sizes; (void)n_in; (void)out_size; (void)ws_size;
  const float* input = (const float*)d_in[0];
  const float* Wd    = (const float*)d_in[1];
  const float* Wf    = (const float*)d_in[2];
  const float* Wpos  = (const float*)d_in[3];
  const float* bpos  = (const float*)d_in[4];
  float4* anchors = (float4*)d_ws;          // B*To*M float4 = 16 KB scratch
  float*  out     = (float*)d_out;

  fps_kernel<<<BB * TO, 32, 0, stream>>>(input, anchors);
  encode_kernel<<<BB * TO * MM, 128, 0, stream>>>(input, Wd, Wf, Wpos, bpos,
                                                  anchors, out);
}